// PratyaharaAttentionBias_17875653886319
// MI455X (gfx1250) — compile-verified
//
#include <hip/hip_runtime.h>
#include <stdint.h>

// out[b,h,i,j] = att[b,h,i,j] + P[idx[b,i], idx[b,j]] * scale[h]
// Memory-bound: ~1.07 GB of streaming traffic -> ~46 us at 23.3 TB/s.

typedef float __attribute__((ext_vector_type(4))) fx4;
typedef int   __attribute__((ext_vector_type(4))) ix4;

#define SEQ            2048
#define HEADS          16
#define VOCAB          256
#define THREADS        256
#define COLS_PER_BLOCK 1024   // 256 threads * 4 cols (one b128 per head)

typedef __attribute__((address_space(1))) int as1_int;
typedef __attribute__((address_space(3))) int as3_int;

__device__ __forceinline__ int clampi(int v, int lo, int hi) {
    return v < lo ? lo : (v > hi ? hi : v);
}

__global__ __launch_bounds__(THREADS)
void pratyahara_bias_kernel(const int*   __restrict__ idx,   // (B, 2048) int32
                            const float* __restrict__ att,   // (B, 16, 2048, 2048)
                            const float* __restrict__ pm,    // (256, 256)
                            const float* __restrict__ scale, // (16,)
                            float*       __restrict__ out)
{
    __shared__ float row[VOCAB];   // P[idx[b,i], :] -- shared by all 16 heads

    const int tid = threadIdx.x;      // 0..255
    const int jt  = blockIdx.x;       // 0..1  (column tile)
    const int i   = blockIdx.y;       // 0..2047
    const int b   = blockIdx.z;       // 0..B-1

    // ---- stage the pratyahara row for idx[b,i] into LDS -------------------
    const int ri = clampi(idx[b * SEQ + i], 0, VOCAB - 1);   // uniform -> s_load
    const float* src = pm + (size_t)ri * VOCAB + tid;        // one float per lane

#if defined(__has_builtin) && __has_builtin(__builtin_amdgcn_global_load_async_to_lds_b32) && __has_builtin(__builtin_amdgcn_s_wait_asynccnt)
    // CDNA5 async copy: VMEM -> LDS without touching VGPRs (ASYNCcnt tracked).
    __builtin_amdgcn_global_load_async_to_lds_b32(
        (as1_int*)src,
        (as3_int*)&row[tid],
        /*offset=*/0, /*cpol=*/0);
    __builtin_amdgcn_s_wait_asynccnt(0);
#else
    row[tid] = *src;   // global_load_b32 + ds_store_b32 fallback
#endif
    __syncthreads();

    // ---- gather this thread's 4 column bias values from LDS ---------------
    const int j0 = jt * COLS_PER_BLOCK + tid * 4;
    const ix4 cj = *(const ix4*)(idx + b * SEQ + j0);        // 16B aligned
    fx4 rel;
    rel.x = row[clampi(cj.x, 0, VOCAB - 1)];
    rel.y = row[clampi(cj.y, 0, VOCAB - 1)];
    rel.z = row[clampi(cj.z, 0, VOCAB - 1)];
    rel.w = row[clampi(cj.w, 0, VOCAB - 1)];

    // ---- stream all 16 heads: NT b128 load, fma, NT b128 store ------------
    size_t off = (size_t)b * HEADS * SEQ * SEQ + (size_t)i * SEQ + j0;
    const size_t hstride = (size_t)SEQ * SEQ;

#pragma unroll
    for (int h = 0; h < HEADS; ++h, off += hstride) {
        const float s = scale[h];                            // uniform -> s_load
        fx4 a = __builtin_nontemporal_load((const fx4*)(att + off));
        fx4 o;
        o.x = fmaf(rel.x, s, a.x);
        o.y = fmaf(rel.y, s, a.y);
        o.z = fmaf(rel.z, s, a.z);
        o.w = fmaf(rel.w, s, a.w);
        __builtin_nontemporal_store(o, (fx4*)(out + off));
    }
}

extern "C" void kernel_launch(void* const* d_in, const int* in_sizes, int n_in,
                              void* d_out, int out_size, void* d_ws, size_t ws_size,
                              hipStream_t stream) {
    const int*   idx = (const int*)  d_in[0];  // phoneme_indices (B, 2048)
    const float* att = (const float*)d_in[1];  // attention_scores (B, 16, 2048, 2048)
    const float* pm  = (const float*)d_in[2];  // pratyahara_matrix (256, 256)
    const float* sc  = (const float*)d_in[3];  // bias_scale (16,)

    const int B = in_sizes[0] / SEQ;           // = 2 for the reference shapes

    dim3 grid(SEQ / COLS_PER_BLOCK, SEQ, B);   // (2, 2048, B) = 8192 blocks
    pratyahara_bias_kernel<<<grid, THREADS, 0, stream>>>(idx, att, pm, sc,
                                                         (float*)d_out);
}